// GCNLayer_26036091748831
// MI455X (gfx1250) — compile-verified
//
#include <hip/hip_runtime.h>

// GCN layer, fused for MI455X (gfx1250, wave32):
//   out = segment_sum(edge_weight * H[edge_src]) @ W + b
// Phase 1 (per wave): deterministic sorted-segment aggregation of 16 node rows
//                     into a private LDS tile (no atomics: edge_dst is sorted).
// Phase 2 (per wave): 16x128 @ 128x128 strip-GEMM with V_WMMA_F32_16X16X4_F32.

#define N_NODES   50000
#define N_EDGES   1600000
#define IN_DIM    128
#define OUT_DIM   128
#define ROW_PAD   132          // 128 + 4 dwords: kills LDS bank conflicts on A reads
#define WPB       8            // waves per block (256 threads, wave32)
#define M_TILES   ((N_NODES + 15) / 16)   // 3125

typedef __attribute__((ext_vector_type(2))) float v2f;
typedef __attribute__((ext_vector_type(8))) float v8f;

__device__ __forceinline__ int lower_bound_i32(const int* __restrict__ a, int n, int key) {
  int lo = 0, hi = n;
  while (lo < hi) {
    int mid = (lo + hi) >> 1;
    if (a[mid] < key) lo = mid + 1; else hi = mid;
  }
  return lo;
}

__global__ __launch_bounds__(256)
void gcn_fused_kernel(const float* __restrict__ H,
                      const int*   __restrict__ esrc,
                      const int*   __restrict__ edst,
                      const float* __restrict__ ew,
                      const float* __restrict__ W,
                      const float* __restrict__ bias,
                      float*       __restrict__ out) {
  __shared__ float smem[WPB * 16 * ROW_PAD];   // 67,584 B

  const int lane   = threadIdx.x & 31;
  const int wib    = threadIdx.x >> 5;                    // wave in block
  const int wave   = blockIdx.x * WPB + wib;              // global wave id
  if (wave >= M_TILES) return;
  const int m0     = wave * 16;                           // first node row
  float* tile = smem + wib * (16 * ROW_PAD);

  // ---------------- Phase 1: aggregation into LDS tile ----------------
  // Lane owns features [4*lane, 4*lane+4). Zero the 16 rows first.
  const int fo = lane * 4;
  {
    const float4 z = make_float4(0.f, 0.f, 0.f, 0.f);
    #pragma unroll
    for (int r = 0; r < 16; ++r)
      *reinterpret_cast<float4*>(tile + r * ROW_PAD + fo) = z;
  }

  // edge_dst is sorted: binary search this wave's contiguous edge range.
  const int e_lo = lower_bound_i32(edst, N_EDGES, m0);
  const int e_hi = lower_bound_i32(edst, N_EDGES, m0 + 16);

  float4 acc = make_float4(0.f, 0.f, 0.f, 0.f);
  int cur = -1;
  for (int e = e_lo; e < e_hi; ++e) {
    const int dst = edst[e];                 // wave-uniform
    if (dst != cur) {
      if (cur >= 0)
        *reinterpret_cast<float4*>(tile + (cur - m0) * ROW_PAD + fo) = acc;
      cur = dst;
      acc = make_float4(0.f, 0.f, 0.f, 0.f);
    }
    const int   src = esrc[e];
    const float w   = ew[e];
    const float4 h  = *reinterpret_cast<const float4*>(H + src * IN_DIM + fo);
    acc.x += w * h.x;  acc.y += w * h.y;
    acc.z += w * h.z;  acc.w += w * h.w;
  }
  if (cur >= 0)
    *reinterpret_cast<float4*>(tile + (cur - m0) * ROW_PAD + fo) = acc;
  // LDS ops are in-order within a wave (DScnt); tile is private to this wave,
  // so no barrier is required before the GEMM phase.

  // ---------------- Phase 2: 16x128 strip GEMM via WMMA ----------------
  // A (16x4 f32) fragment layout: lanes 0-15 -> M=lane, K={k,k+1};
  //                               lanes 16-31 -> M=lane-16, K={k+2,k+3}.
  // B (4x16 f32) fragment mirrors the K split; N = lane & 15.
  const int l15 = lane & 15;
  const int hi2 = (lane >> 4) * 2;      // 0 or 2: K sub-offset for this half-wave

  v8f d[8];
  #pragma unroll
  for (int t = 0; t < 8; ++t) d[t] = (v8f){0.f,0.f,0.f,0.f,0.f,0.f,0.f,0.f};

  const float* arow = tile + l15 * ROW_PAD + hi2;  // this lane's A row base

  for (int k = 0; k < IN_DIM; k += 4) {
    const v2f a = *reinterpret_cast<const v2f*>(arow + k);   // ds_load_b64
    #pragma unroll
    for (int t = 0; t < 8; ++t) {
      const float* bp = W + (k + hi2) * OUT_DIM + t * 16 + l15;
      v2f bfrag;
      bfrag.x = bp[0];            // B[k+hi2  ][n]  (coalesced across lanes)
      bfrag.y = bp[OUT_DIM];      // B[k+hi2+1][n]
      d[t] = __builtin_amdgcn_wmma_f32_16x16x4_f32(
          /*neg_a=*/false, a, /*neg_b=*/false, bfrag,
          /*c_mod=*/(short)0, d[t], /*reuse_a=*/false, /*reuse_b=*/false);
    }
  }

  // Epilogue: C/D layout -> VGPR v holds row m0 + 8*(lane>=16) + v, col l15.
  const int row_base = m0 + (lane >> 4) * 8;
  #pragma unroll
  for (int t = 0; t < 8; ++t) {
    const int   n  = t * 16 + l15;
    const float bv = bias[n];
    float* op = out + row_base * OUT_DIM + n;
    #pragma unroll
    for (int v = 0; v < 8; ++v)
      op[v * OUT_DIM] = d[t][v] + bv;
  }
}

extern "C" void kernel_launch(void* const* d_in, const int* in_sizes, int n_in,
                              void* d_out, int out_size, void* d_ws, size_t ws_size,
                              hipStream_t stream) {
  const float* H    = (const float*)d_in[0];
  const int*   esrc = (const int*)  d_in[1];
  const int*   edst = (const int*)  d_in[2];
  const float* ew   = (const float*)d_in[3];
  const float* W    = (const float*)d_in[4];
  const float* bias = (const float*)d_in[5];
  float*       out  = (float*)d_out;

  const int blocks = (M_TILES + WPB - 1) / WPB;   // 391
  gcn_fused_kernel<<<blocks, 256, 0, stream>>>(H, esrc, edst, ew, W, bias, out);
}